// Deeds_14577119002782
// MI455X (gfx1250) — compile-verified
//
#include <hip/hip_runtime.h>

// ---------------------------------------------------------------------------
// deeds cost-volume pipeline for MI455X (gfx1250, wave32)
//   K0 transFeat : [C][z][y][x] -> [z][y][x][C]  (float4-friendly sampling)
//   K1 costK     : build dc[G^3, D^3]  (one WGP workgroup per grid point)
//   K2 minavgK   : pad3->min3->avg3->avg3 fully in LDS per grid point
//                  (separable clamped min + separable [1,2,3,2,1]/9 triangle)
//   K3 transposeK: tiled 32x32 layout flips between g-major and d-major
//   K4 avgGridK  : pad2->avg3->avg3 == clamped 5-tap triangle per axis,
//                  done as GEMM  out = T29 x in  with V_WMMA_F32_16X16X4_F32
//                  (boundary clamp folded into dense banded operator T29)
//   K5 softpredK : fused softmax over D^3 + soft-argmax (cost_soft @ shift)
// ---------------------------------------------------------------------------

typedef float v2f __attribute__((ext_vector_type(2)));
typedef float v8f __attribute__((ext_vector_type(8)));

constexpr int GN = 29;                 // grid size
constexpr int DN = 15;                 // displacement width
constexpr int SN = 48;                 // feature spatial size
constexpr int G3 = GN * GN * GN;       // 24389
constexpr int D3 = DN * DN * DN;       // 3375
constexpr size_t NTOT = (size_t)G3 * D3;
constexpr size_t FTSZ = (size_t)SN * SN * SN * 16;  // 1,769,472 floats

__device__ __forceinline__ int clampi(int v, int lo, int hi) {
    return v < lo ? lo : (v > hi ? hi : v);
}
__device__ __forceinline__ float gridc(int i) {            // affine_grid coord
    return (2.f * i + 1.f) / 29.f - 1.f;
}
__device__ __forceinline__ float shiftc(int i) {           // disp coord (R=0.4)
    return 0.4f * ((2.f * i + 1.f) / 15.f - 1.f);
}
__device__ __forceinline__ float unnorm48(float c) {       // border clamp
    float v = ((c + 1.f) * 48.f - 1.f) * 0.5f;
    return fminf(fmaxf(v, 0.f), 47.f);
}

// ---------------- trilinear 16-channel sampler on [z][y][x][16] ------------
__device__ __forceinline__ void addCorner(const float4* __restrict__ ft,
                                          int zi, int yi, int xi, float w,
                                          float o[16]) {
    const float4* p = ft + ((((size_t)zi * 48 + yi) * 48 + xi) << 2);
#pragma unroll
    for (int j = 0; j < 4; ++j) {
        float4 v = p[j];
        o[4 * j + 0] = fmaf(w, v.x, o[4 * j + 0]);
        o[4 * j + 1] = fmaf(w, v.y, o[4 * j + 1]);
        o[4 * j + 2] = fmaf(w, v.z, o[4 * j + 2]);
        o[4 * j + 3] = fmaf(w, v.w, o[4 * j + 3]);
    }
}

__device__ __forceinline__ void tri16(const float4* __restrict__ ft,
                                      float px, float py, float pz,
                                      float o[16]) {
    int x0 = (int)px, y0 = (int)py, z0 = (int)pz;       // px,py,pz in [0,47]
    int x1 = min(x0 + 1, 47), y1 = min(y0 + 1, 47), z1 = min(z0 + 1, 47);
    float wx = px - x0, wy = py - y0, wz = pz - z0;
    float ux = 1.f - wx, uy = 1.f - wy, uz = 1.f - wz;
#pragma unroll
    for (int i = 0; i < 16; ++i) o[i] = 0.f;
    addCorner(ft, z0, y0, x0, uz * uy * ux, o);
    addCorner(ft, z0, y0, x1, uz * uy * wx, o);
    addCorner(ft, z0, y1, x0, uz * wy * ux, o);
    addCorner(ft, z0, y1, x1, uz * wy * wx, o);
    addCorner(ft, z1, y0, x0, wz * uy * ux, o);
    addCorner(ft, z1, y0, x1, wz * uy * wx, o);
    addCorner(ft, z1, y1, x0, wz * wy * ux, o);
    addCorner(ft, z1, y1, x1, wz * wy * wx, o);
}

// ---------------- K0: channel-major -> voxel-major feature transpose -------
__global__ void transFeat(const float* __restrict__ f, float* __restrict__ ft) {
    __shared__ float t[768];                        // 16 ch x 48 x
    int zy = blockIdx.x;                            // z*48 + y
    for (int i = threadIdx.x; i < 768; i += 256) {
        int c = i / 48, x = i % 48;                 // coalesced x-runs
        t[i] = f[(size_t)c * 110592 + (size_t)zy * 48 + x];
    }
    __syncthreads();
    for (int i = threadIdx.x; i < 768; i += 256) {
        int x = i / 16, c = i % 16;                 // coalesced writes
        ft[((size_t)zy * 48 + x) * 16 + c] = t[c * 48 + x];
    }
}

// ---------------- K1: cost volume build ------------------------------------
__global__ void costK(const float4* __restrict__ f0t,
                      const float4* __restrict__ f5t,
                      const float* __restrict__ alpha,
                      float* __restrict__ dc) {
    int g = blockIdx.x;
    int gx = g % 29, gy = (g / 29) % 29, gz = g / 841;
    float cgx = gridc(gx), cgy = gridc(gy), cgz = gridc(gz);

    __shared__ float fixS[16];
    if (threadIdx.x == 0) {
        float o[16];
        tri16(f0t, unnorm48(cgx), unnorm48(cgy), unnorm48(cgz), o);
#pragma unroll
        for (int i = 0; i < 16; ++i) fixS[i] = o[i];
    }
    __syncthreads();
    float fr[16];
#pragma unroll
    for (int i = 0; i < 16; ++i) fr[i] = fixS[i];

    float a0 = alpha[0], a1 = alpha[1];
    size_t base = (size_t)g * D3;
    for (int s = threadIdx.x; s < D3; s += 256) {
        int sx = s % 15, sy = (s / 15) % 15, sz = s / 225;
        float m[16];
        tri16(f5t, unnorm48(cgx + shiftc(sx)), unnorm48(cgy + shiftc(sy)),
              unnorm48(cgz + shiftc(sz)), m);
        float acc = 0.f;
#pragma unroll
        for (int i = 0; i < 16; ++i) {
            float d = fr[i] - m[i];
            acc = fmaf(d, d, acc);
        }
        dc[base + s] = a1 + a0 * acc;
    }
}

// ---------------- K2: minavg_disp, fully LDS-resident per grid point -------
// mode 0: v = in1 ; mode 1: v = alpha[4] + alpha[2]*in1 + alpha[3]*in2
__global__ void minavgK(const float* __restrict__ in1,
                        const float* __restrict__ in2,
                        float* __restrict__ out,
                        const float* __restrict__ alpha, int mode) {
    extern __shared__ float sm2[];
    float* xv = sm2;                // 3375
    float* b1 = sm2 + 3375;        // 6859
    float* b2 = b1 + 6859;         // 6859
    int g = blockIdx.x, tid = threadIdx.x;
    size_t base = (size_t)g * D3;

    if (mode) {
        float k4 = alpha[4], k2 = alpha[2], k3 = alpha[3];
        for (int i = tid; i < 3375; i += 256)
            xv[i] = k4 + k2 * in1[base + i] + k3 * in2[base + i];
    } else {
        for (int i = tid; i < 3375; i += 256) xv[i] = in1[base + i];
    }
    __syncthreads();
    // clamped min along fastest dim: 15 -> 19 extended
    for (int e = tid; e < 4275; e += 256) {
        int u = e % 19, ab = e / 19;
        const float* r = xv + ab * 15;
        float m = r[clampi(u - 3, 0, 14)];
        m = fminf(m, r[clampi(u - 2, 0, 14)]);
        m = fminf(m, r[clampi(u - 1, 0, 14)]);
        b1[e] = m;
    }
    __syncthreads();
    // clamped min along middle dim
    for (int e = tid; e < 5415; e += 256) {
        int u = e % 19, v = (e / 19) % 19, a = e / 361;
        float m = b1[(a * 15 + clampi(v - 3, 0, 14)) * 19 + u];
        m = fminf(m, b1[(a * 15 + clampi(v - 2, 0, 14)) * 19 + u]);
        m = fminf(m, b1[(a * 15 + clampi(v - 1, 0, 14)) * 19 + u]);
        b2[e] = m;
    }
    __syncthreads();
    // clamped min along slowest dim -> M (19^3)
    for (int e = tid; e < 6859; e += 256) {
        int u = e % 19, v = (e / 19) % 19, w = e / 361;
        float m = b2[(clampi(w - 3, 0, 14) * 19 + v) * 19 + u];
        m = fminf(m, b2[(clampi(w - 2, 0, 14) * 19 + v) * 19 + u]);
        m = fminf(m, b2[(clampi(w - 1, 0, 14) * 19 + v) * 19 + u]);
        b1[e] = m;
    }
    __syncthreads();
    const float t0 = 1.f / 9.f, t1 = 2.f / 9.f, t2 = 3.f / 9.f;
    // avg3∘avg3 == valid 5-tap triangle: along u (19 -> 15)
    for (int e = tid; e < 5415; e += 256) {
        int i = e % 15, wv_ = e / 15;
        const float* r = b1 + wv_ * 19 + i;
        b2[e] = t0 * r[0] + t1 * r[1] + t2 * r[2] + t1 * r[3] + t0 * r[4];
    }
    __syncthreads();
    // along v
    for (int e = tid; e < 4275; e += 256) {
        int i = e % 15, j = (e / 15) % 15, w = e / 225;
        const float* r = b2 + (w * 19 + j) * 15 + i;
        b1[e] = t0 * r[0] + t1 * r[15] + t2 * r[30] + t1 * r[45] + t0 * r[60];
    }
    __syncthreads();
    // along w + store
    for (int e = tid; e < 3375; e += 256) {
        int i = e % 15, j = (e / 15) % 15, k = e / 225;
        const float* r = b1 + (k * 15 + j) * 15 + i;
        out[base + e] =
            t0 * r[0] + t1 * r[225] + t2 * r[450] + t1 * r[675] + t0 * r[900];
    }
}

// ---------------- K3: tiled transpose [R,C] -> [C,R] -----------------------
__global__ void transposeK(const float* __restrict__ in, float* __restrict__ out,
                           int R, int Cc) {
    __shared__ float tile[32][33];
    int cb = blockIdx.x * 32, rb = blockIdx.y * 32;
    int tx = threadIdx.x, ty = threadIdx.y;              // 32 x 8
    for (int i = ty; i < 32; i += 8) {
        int r = rb + i, c = cb + tx;
        if (r < R && c < Cc) tile[i][tx] = in[(size_t)r * Cc + c];
    }
    __syncthreads();
    for (int i = ty; i < 32; i += 8) {
        int c = cb + i, r = rb + tx;
        if (r < R && c < Cc) out[(size_t)c * R + r] = tile[tx][i];
    }
}

// ---------------- K4: avg_grid as WMMA f32 GEMM per axis -------------------
// Dense banded operator: out[i] = sum_k w[k] * in[clamp(i+k,0,28)],
// w = [1,2,3,2,1]/9.  T folded with replicate-pad boundary.
__device__ __forceinline__ float triW29(int i, int j) {
    if (i > 28 || j > 28) return 0.f;
    float s = 0.f;
#pragma unroll
    for (int k = -2; k <= 2; ++k) {
        int c = i + k;
        c = c < 0 ? 0 : (c > 28 ? 28 : c);
        float w = (k == 0) ? (3.f / 9.f)
                           : ((k == -1 || k == 1) ? (2.f / 9.f) : (1.f / 9.f));
        if (c == j) s += w;
    }
    return s;
}

__global__ void avgGridK(const float* __restrict__ in, float* __restrict__ out) {
    extern __shared__ float sm4[];
    constexpr int BUF = 24448;                  // 29^3 padded
    float* s0 = sm4;
    float* s1 = sm4 + BUF;
    int d = blockIdx.x, tid = threadIdx.x;
    for (int i = tid; i < G3; i += 256) s0[i] = in[(size_t)d * G3 + i];
    __syncthreads();

    const int sfv[3] = {1, 29, 841};            // filtered-axis stride
    const int sAv[3] = {841, 841, 29};          // column decode strides
    const int sBv[3] = {29, 1, 1};
    float* bin = s0;
    float* bout = s1;
    int lane = tid & 31, wv = tid >> 5;
    int ln = lane & 15;
    int hi = (lane >> 4) & 1;                   // wave32 half select

    for (int p = 0; p < 3; ++p) {
        int sf = sfv[p], sA = sAv[p], sB = sBv[p];
        // 2 M-tiles (29 rows) x 53 N-tiles (841 cols), wave-uniform loop
        for (int t = wv; t < 106; t += 8) {
            int mt = t & 1, nt = t >> 1;
            int m = mt * 16 + ln;               // output row of T (0..31)
            int n = nt * 16 + ln;               // column (0..847)
            int ncl = n > 840 ? 840 : n;
            int colb = (ncl / 29) * sA + (ncl % 29) * sB;
            v8f acc = {0.f, 0.f, 0.f, 0.f, 0.f, 0.f, 0.f, 0.f};
#pragma unroll
            for (int k0 = 0; k0 < 32; k0 += 4) {    // K = 29 padded to 32
                int kb = k0 + (hi << 1);
                v2f a, b;
                a.x = triW29(m, kb);                 // A 16x4: lanes 0-15 K0/K1,
                a.y = triW29(m, kb + 1);             //          lanes16-31 K2/K3
                b.x = (kb < 29) ? bin[kb * sf + colb] : 0.f;
                b.y = (kb < 28) ? bin[(kb + 1) * sf + colb] : 0.f;
                acc = __builtin_amdgcn_wmma_f32_16x16x4_f32(
                    false, a, false, b, (short)0, acc, false, false);
            }
            if (n < 841) {                           // D: vgpr j -> row j(+8)
                int rb = mt * 16 + (hi << 3);
#pragma unroll
                for (int j = 0; j < 8; ++j) {
                    int row = rb + j;
                    if (row < 29) bout[row * sf + colb] = acc[j];
                }
            }
        }
        __syncthreads();
        float* tp = bin; bin = bout; bout = tp;      // ping-pong
    }
    for (int i = tid; i < G3; i += 256) out[(size_t)d * G3 + i] = bin[i];
}

// ---------------- K5: fused softmax + soft-argmax --------------------------
__global__ void softpredK(float* __restrict__ buf, float* __restrict__ pred,
                          const float* __restrict__ alpha) {
    int g = blockIdx.x, tid = threadIdx.x;
    size_t base = (size_t)g * D3;
    float a5 = alpha[5];
    float loc[14];
    int cnt = 0;
    float mx = -3.4e38f;
    for (int s = tid; s < D3; s += 256) {
        float L = -a5 * buf[base + s];
        loc[cnt++] = L;
        mx = fmaxf(mx, L);
    }
    __shared__ float r0[256], r1[256], r2[256], r3[256];
    r0[tid] = mx;
    __syncthreads();
    for (int o = 128; o > 0; o >>= 1) {
        if (tid < o) r0[tid] = fmaxf(r0[tid], r0[tid + o]);
        __syncthreads();
    }
    float M = r0[0];
    __syncthreads();
    float sum = 0.f, px = 0.f, py = 0.f, pz = 0.f;
    cnt = 0;
    for (int s = tid; s < D3; s += 256) {
        float e = expf(loc[cnt] - M);
        loc[cnt++] = e;
        int sx = s % 15, sy = (s / 15) % 15, sz = s / 225;
        sum += e;
        px = fmaf(e, shiftc(sx), px);
        py = fmaf(e, shiftc(sy), py);
        pz = fmaf(e, shiftc(sz), pz);
    }
    r0[tid] = sum; r1[tid] = px; r2[tid] = py; r3[tid] = pz;
    __syncthreads();
    for (int o = 128; o > 0; o >>= 1) {
        if (tid < o) {
            r0[tid] += r0[tid + o]; r1[tid] += r1[tid + o];
            r2[tid] += r2[tid + o]; r3[tid] += r3[tid + o];
        }
        __syncthreads();
    }
    float inv = 1.f / r0[0];
    cnt = 0;
    for (int s = tid; s < D3; s += 256) buf[base + s] = loc[cnt++] * inv;
    if (tid == 0) {
        pred[(size_t)g * 3 + 0] = r1[0] * inv;
        pred[(size_t)g * 3 + 1] = r2[0] * inv;
        pred[(size_t)g * 3 + 2] = r3[0] * inv;
    }
}

// ---------------------------------------------------------------------------
extern "C" void kernel_launch(void* const* d_in, const int* in_sizes, int n_in,
                              void* d_out, int out_size, void* d_ws, size_t ws_size,
                              hipStream_t stream) {
    (void)in_sizes; (void)n_in; (void)out_size; (void)ws_size;
    const float* feat00 = (const float*)d_in[0];
    const float* feat50 = (const float*)d_in[1];
    const float* alpha  = (const float*)d_in[2];

    float* ws  = (float*)d_ws;
    float* f0t = ws;                       // 1,769,472 floats (16B aligned)
    float* f5t = ws + FTSZ;
    float* A   = ws + 2 * FTSZ;            // big buffer #1 (holds dc)
    float* B   = A + NTOT;                 // big buffer #2
    float* O   = (float*)d_out;            // big buffer #3 == cost_soft area
    float* pred = O + NTOT;                // pred_xyz tail of d_out

    dim3 tb(32, 8);
    size_t smK2 = (size_t)(3375 + 2 * 6859) * sizeof(float);   // ~68 KB
    size_t smK4 = (size_t)2 * 24448 * sizeof(float);           // ~195 KB (<320 KB WGP LDS)

    // features -> voxel-major
    transFeat<<<2304, 256, 0, stream>>>(feat00, f0t);
    transFeat<<<2304, 256, 0, stream>>>(feat50, f5t);
    // dc
    costK<<<G3, 256, 0, stream>>>((const float4*)f0t, (const float4*)f5t, alpha, A);
    // c = minavg_disp(dc)
    minavgK<<<G3, 256, smK2, stream>>>(A, A, B, alpha, 0);
    // ca = avg_grid over grid dims (transpose -> filter -> transpose)
    transposeK<<<dim3(106, 763), tb, 0, stream>>>(B, O, G3, D3);
    avgGridK<<<D3, 256, smK4, stream>>>(O, B);
    transposeK<<<dim3(763, 106), tb, 0, stream>>>(B, O, D3, G3);
    // c = minavg_disp(alpha4 + alpha2*dc + alpha3*ca)   (affine fused in load)
    minavgK<<<G3, 256, smK2, stream>>>(A, O, B, alpha, 1);
    // ca2 = avg_grid(...)
    transposeK<<<dim3(106, 763), tb, 0, stream>>>(B, A, G3, D3);
    avgGridK<<<D3, 256, smK4, stream>>>(A, B);
    transposeK<<<dim3(763, 106), tb, 0, stream>>>(B, O, D3, G3);
    // softmax + soft-argmax, in place on d_out
    softpredK<<<G3, 256, 0, stream>>>(O, pred, alpha);
}